// TemporalConstraintGraph_50276887167212
// MI455X (gfx1250) — compile-verified
//
#include <hip/hip_runtime.h>
#include <hip/hip_bf16.h>

typedef float v2f __attribute__((ext_vector_type(2)));
typedef float v8f __attribute__((ext_vector_type(8)));

#define FEAT 600
#define HALF_FEAT 300
#define PAIRS_PER_WAVE 16
#define WAVES_PER_BLOCK 8
#define BLOCK_THREADS (WAVES_PER_BLOCK * 32)
#define PAIRS_PER_BLOCK (WAVES_PER_BLOCK * PAIRS_PER_WAVE)

__global__ void tcg_zero_out(float* out) { out[0] = 0.0f; }

__global__ __launch_bounds__(BLOCK_THREADS)
void tcg_hinge_kernel(const float* __restrict__ embs,
                      const int* __restrict__ pairs,   // (n_pairs, 2) flat
                      float* __restrict__ out,
                      int n_pairs, float inv_n) {
    const int lane = threadIdx.x & 31;
    const int wave = threadIdx.x >> 5;
    const int pi   = lane & 15;   // pair slot within the wave's 16-pair group
    const int half = lane >> 4;   // which half of the feature vector this lane sums

    const long long base = ((long long)blockIdx.x * WAVES_PER_BLOCK + wave)
                         * (long long)PAIRS_PER_WAVE;
    const long long pairIdx = base + pi;
    const bool valid = (pairIdx < (long long)n_pairs);
    const long long safe = valid ? pairIdx : 0;

    const int ri = pairs[2 * safe + 0];
    const int rj = pairs[2 * safe + 1];

    const float* __restrict__ fi = embs + (long long)ri * FEAT + half * HALF_FEAT;
    const float* __restrict__ fj = embs + (long long)rj * FEAT + half * HALF_FEAT;

    // Per-lane streaming gather: 75 aligned float4 (b128) loads per row half.
    // Four accumulators -> compiler packs into v_pk_fma_f32 chains.
    float a0 = 0.f, a1 = 0.f, a2 = 0.f, a3 = 0.f;
    #pragma unroll 5
    for (int t = 0; t < HALF_FEAT; t += 4) {
        const float4 va = *reinterpret_cast<const float4*>(fi + t);
        const float4 vb = *reinterpret_cast<const float4*>(fj + t);
        const float d0 = va.x - vb.x;
        const float d1 = va.y - vb.y;
        const float d2 = va.z - vb.z;
        const float d3 = va.w - vb.w;
        a0 = fmaf(d0, d0, a0);
        a1 = fmaf(d1, d1, a1);
        a2 = fmaf(d2, d2, a2);
        a3 = fmaf(d3, d3, a3);
    }
    float acc = (a0 + a1) + (a2 + a3);
    acc = valid ? acc : 0.0f;          // select, keeps EXEC all-1s for WMMA

    // Cross-lane per-pair reduction on the matrix pipe:
    // A (16x4 f32): row m = pair m's two half-row partials (lane m, lane m+16).
    // B (4x16) = ones  =>  D[m][n] = ||fi-fj||^2 for pair m, replicated over n.
    v2f A; A[0] = acc; A[1] = 0.0f;
    v2f B; B[0] = 1.0f; B[1] = 1.0f;
    v8f C = {};
    C = __builtin_amdgcn_wmma_f32_16x16x4_f32(
            /*neg_a=*/false, A, /*neg_b=*/false, B,
            /*c_mod=*/(short)0, C, /*reuse_a=*/false, /*reuse_b=*/false);

    // Lanes 0 and 16 hold pairs base+0..7 and base+8..15 in C[0..7].
    // Branchless epilogue: raw v_sqrt_f32 + cndmask per element.
    float local = 0.0f;
    if ((lane & 15) == 0) {
        const long long pbase = base + half * 8;
        #pragma unroll
        for (int r = 0; r < 8; ++r) {
            const float dist = __builtin_amdgcn_sqrtf(C[r]);
            const float h = fmaxf(1.0f - dist, 0.0f);
            local += (pbase + r < (long long)n_pairs) ? h : 0.0f;
        }
    }

    __shared__ float s_part[WAVES_PER_BLOCK * 2];
    if ((lane & 15) == 0) s_part[wave * 2 + half] = local;
    __syncthreads();

    if (threadIdx.x == 0) {
        float t = 0.0f;
        #pragma unroll
        for (int k = 0; k < WAVES_PER_BLOCK * 2; ++k) t += s_part[k];
        atomicAdd(out, t * inv_n);
    }
}

extern "C" void kernel_launch(void* const* d_in, const int* in_sizes, int n_in,
                              void* d_out, int out_size, void* d_ws, size_t ws_size,
                              hipStream_t stream) {
    const float* embs  = (const float*)d_in[0];
    const int*   pairs = (const int*)d_in[1];
    float*       out   = (float*)d_out;

    const int n_pairs = in_sizes[1] / 2;
    const float inv_n = 1.0f / (float)(n_pairs > 0 ? n_pairs : 1);

    tcg_zero_out<<<1, 1, 0, stream>>>(out);

    const long long nblocks =
        ((long long)n_pairs + PAIRS_PER_BLOCK - 1) / PAIRS_PER_BLOCK;
    tcg_hinge_kernel<<<dim3((unsigned)nblocks), dim3(BLOCK_THREADS), 0, stream>>>(
        embs, pairs, out, n_pairs, inv_n);
}